// Attention_30245159698745
// MI455X (gfx1250) — compile-verified
//
#include <hip/hip_runtime.h>
#include <math.h>

typedef __attribute__((ext_vector_type(2))) float v2f;
typedef __attribute__((ext_vector_type(4))) float v4f;
typedef __attribute__((ext_vector_type(8))) float v8f;
typedef __attribute__((ext_vector_type(4))) int   v4i;

#define DIM     4096
#define NHEADS  32
#define NKV     8
#define HD      128
#define SEQ     1024
#define BATCH   4
#define MROWS   (BATCH * SEQ)   // 4096 rows of x / q / attn-out

// --------------------------------------------------------------------------
// CDNA5 async global->LDS copy (ASYNCcnt-tracked), with portable fallback.
// Builtin signature (probe-confirmed via diagnostics): (v4i __device__* src,
// v4i __shared__* dst, imm offset, imm cpol) — non-const src.
// --------------------------------------------------------------------------
#if defined(__gfx1250__) && __has_builtin(__builtin_amdgcn_global_load_async_to_lds_b128)
#define HAVE_ASYNC_LDS 1
typedef __attribute__((address_space(1))) v4i gv4i;
typedef __attribute__((address_space(3))) v4i lv4i;
static __device__ __forceinline__ void async_copy16(const float* g, float* l) {
  __builtin_amdgcn_global_load_async_to_lds_b128((gv4i*)g, (lv4i*)l, 0, 0);
}
#else
#define HAVE_ASYNC_LDS 0
static __device__ __forceinline__ void async_copy16(const float* g, float* l) {
  *(v4f*)l = *(const v4f*)g;
}
#endif

static __device__ __forceinline__ void stage_wait() {
#if HAVE_ASYNC_LDS
  asm volatile("s_wait_asynccnt 0" ::: "memory");
#endif
  __syncthreads();
}

// Exact fp32 matrix op: D(16x16,f32) = A(16x4,f32) * B(4x16,f32) + C
static __device__ __forceinline__ v8f wmma_f32(v2f a, v2f b, v8f c) {
  return __builtin_amdgcn_wmma_f32_16x16x4_f32(
      /*neg_a=*/false, a, /*neg_b=*/false, b,
      /*c_mod=*/(short)0, c, /*reuse_a=*/false, /*reuse_b=*/false);
}

// ---------------------------------------------------------------------------
// GEMM: C[M,N] = A[M,K] * B[K,N], fp32 row-major.
// 256 threads = 8 wave32. Workgroup tile 128(M) x 128(N) x 16(K), double-
// buffered LDS staged via async global->LDS b128 copies.
// Wave tile 32x64 (2x4 accumulator grid) to maximize WMMA per DS fragment.
// Fragment layouts per the ISA 16x4 / 4x16 fp32 WMMA tables:
//   A: lane(half,hi): a.x=A[m][k+2hi],  a.y=A[m][k+2hi+1]
//   B: lane(half,hi): b.x=B[k+2hi][n],  b.y=B[k+2hi+1][n]
//   C: vgpr r, lane(half,hi) -> element (r+8hi, half)
// ---------------------------------------------------------------------------
#define BM   128
#define BN   128
#define BK   16
#define APAD 20   // A tile row stride in floats (16B-aligned chunks, 16 banks)

__global__ __launch_bounds__(256) void gemm_wmma_f32(
    const float* __restrict__ A, const float* __restrict__ B,
    float* __restrict__ C, int M, int N, int K) {
  __shared__ float Asm[2][BM * APAD];  // 2 x 10240 B
  __shared__ float Bsm[2][BK * BN];    // 2 x  8192 B

  const int tid  = threadIdx.x;
  const int lane = tid & 31;
  const int wave = tid >> 5;
  const int half = lane & 15;
  const int hi   = lane >> 4;
  const int wm = wave >> 1;            // 0..3  (row block of 32)
  const int wn = wave & 1;             // 0..1  (col block of 64)
  const int blockM = blockIdx.y * BM;
  const int blockN = blockIdx.x * BN;

  // Stage one BMxBK A tile + BKxBN B tile into LDS buffer `buf`.
  auto stage = [&](int buf, int kt) {
    const int k0 = kt * BK;
    // A: 512 chunks of 4 floats, 2 per thread
#pragma unroll
    for (int i = 0; i < 2; ++i) {
      const int c = tid + 256 * i;
      const int m = c >> 2;
      const int j = (c & 3) * 4;
      async_copy16(A + (long)(blockM + m) * K + k0 + j,
                   &Asm[buf][m * APAD + j]);
    }
    // B: 512 chunks of 4 floats, 2 per thread
#pragma unroll
    for (int i = 0; i < 2; ++i) {
      const int c = tid + 256 * i;
      const int k = c >> 5;            // 32 chunks per 128-float row
      const int j = (c & 31) * 4;
      async_copy16(B + (long)(k0 + k) * N + blockN + j,
                   &Bsm[buf][k * BN + j]);
    }
  };

  v8f acc[2][4];
#pragma unroll
  for (int i = 0; i < 2; ++i)
#pragma unroll
    for (int f = 0; f < 4; ++f) acc[i][f] = (v8f){};

  const int NT = K / BK;

  stage(0, 0);
  for (int kt = 0; kt < NT; ++kt) {
    const int buf = kt & 1;
    stage_wait();                        // own asyncs done + all waves arrived
    if (kt + 1 < NT) stage(buf ^ 1, kt + 1);

    const float* As = Asm[buf];
    const float* Bs = Bsm[buf];
    const int n0w = wn * 64;
#pragma unroll
    for (int ks = 0; ks < BK; ks += 4) {
      const int kk = ks + 2 * hi;
      v2f a0 = *(const v2f*)&As[(wm * 32 + half) * APAD + kk];
      v2f a1 = *(const v2f*)&As[(wm * 32 + 16 + half) * APAD + kk];
      v2f b[4];
#pragma unroll
      for (int f = 0; f < 4; ++f) {
        b[f].x = Bs[kk * BN + n0w + 16 * f + half];
        b[f].y = Bs[(kk + 1) * BN + n0w + 16 * f + half];
      }
#pragma unroll
      for (int f = 0; f < 4; ++f) {
        acc[0][f] = wmma_f32(a0, b[f], acc[0][f]);
        acc[1][f] = wmma_f32(a1, b[f], acc[1][f]);
      }
    }
    __syncthreads();                     // tile fully read before re-staging
  }

  const int m0 = blockM + wm * 32;
  const int n0 = blockN + wn * 64;
#pragma unroll
  for (int i = 0; i < 2; ++i) {
#pragma unroll
    for (int f = 0; f < 4; ++f) {
#pragma unroll
      for (int r = 0; r < 8; ++r) {
        const long row = m0 + 16 * i + r + 8 * hi;
        C[row * N + n0 + 16 * f + half] = acc[i][f][r];
      }
    }
  }
}

// ---------------------------------------------------------------------------
// RoPE, applied in place to [MROWS, nheads, HD] (interleaved even/odd pairs).
// ---------------------------------------------------------------------------
__global__ void rope_kernel(float* __restrict__ X, int nheads, long total) {
  long idx = (long)blockIdx.x * blockDim.x + threadIdx.x;
  if (idx >= total) return;
  const int dpair = (int)(idx & 63);
  long t = idx >> 6;
  const int hIdx = (int)(t % nheads);
  const long row = t / nheads;           // b*SEQ + s
  const int s = (int)(row & (SEQ - 1));

  const float inv = powf(10000.0f, -(float)(2 * dpair) / (float)HD);
  const float ang = (float)s * inv;
  float sn, cs;
  sincosf(ang, &sn, &cs);

  float* p = X + (row * nheads + hIdx) * HD + 2 * dpair;
  const float xr = p[0];
  const float xi = p[1];
  p[0] = xr * cs - xi * sn;
  p[1] = xr * sn + xi * cs;
}

// half-wave (16-lane) row reductions for the 16x16 C layout
static __device__ __forceinline__ float halfmax(float v) {
  v = fmaxf(v, __shfl_xor(v, 1, 32));
  v = fmaxf(v, __shfl_xor(v, 2, 32));
  v = fmaxf(v, __shfl_xor(v, 4, 32));
  v = fmaxf(v, __shfl_xor(v, 8, 32));
  return v;
}
static __device__ __forceinline__ float halfsum(float v) {
  v += __shfl_xor(v, 1, 32);
  v += __shfl_xor(v, 2, 32);
  v += __shfl_xor(v, 4, 32);
  v += __shfl_xor(v, 8, 32);
  return v;
}

// ---------------------------------------------------------------------------
// Flash-style causal GQA attention. One wave handles one (b, h, 16-row qblock).
// No block barriers (waves have different trip counts); per-wave LDS transpose
// buffer synchronized with in-order DS + s_wait_dscnt.
// ---------------------------------------------------------------------------
__global__ __launch_bounds__(256) void attn_fa(
    const float* __restrict__ Q, const float* __restrict__ K,
    const float* __restrict__ V, float* __restrict__ O) {
  __shared__ float pbuf[8][16 * 17];

  const int lane = threadIdx.x & 31;
  const int wave = threadIdx.x >> 5;
  const int half = lane & 15;
  const int hi   = lane >> 4;

  const int gwave = blockIdx.x * 8 + wave;       // 0 .. 8191
  const int qblk  = gwave & 63;                  // SEQ/16 = 64
  const int h     = (gwave >> 6) & 31;
  const int b     = gwave >> 11;
  const int kvh   = h >> 2;                      // N_REP = 4
  const int qBase = qblk * 16;

  const float* Qp = Q + ((long)b * SEQ * NHEADS + h) * HD;
  const float* Kp = K + ((long)b * SEQ * NKV + kvh) * HD;
  const float* Vp = V + ((long)b * SEQ * NKV + kvh) * HD;

  v2f qa[32];
  {
    const float* qrow = Qp + (long)(qBase + half) * (NHEADS * HD) + 2 * hi;
#pragma unroll
    for (int kc = 0; kc < 32; ++kc) qa[kc] = *(const v2f*)(qrow + 4 * kc);
  }

  v8f o[8];
#pragma unroll
  for (int f = 0; f < 8; ++f) o[f] = (v8f){};
  float mrun[8], lrun[8];
#pragma unroll
  for (int r = 0; r < 8; ++r) { mrun[r] = -INFINITY; lrun[r] = 0.0f; }

  const float scale = 0.08838834764831845f;      // 1/sqrt(128)

  for (int kb = 0; kb <= qblk; ++kb) {
    const int kBase = kb * 16;

    // S = Q * K^T (16x16)
    v8f st = {};
    const float* krow = Kp + (long)(kBase + half) * (NKV * HD) + 2 * hi;
#pragma unroll
    for (int kc = 0; kc < 32; ++kc) {
      v2f bf = *(const v2f*)(krow + 4 * kc);
      st = wmma_f32(qa[kc], bf, st);
    }

    const bool diag = (kb == qblk);
    float p[8];
#pragma unroll
    for (int r = 0; r < 8; ++r) {
      float v = st[r] * scale;
      if (diag && (kBase + half > qBase + r + 8 * hi)) v = -INFINITY;
      p[r] = v;
    }
#pragma unroll
    for (int r = 0; r < 8; ++r) {
      const float mnew  = fmaxf(mrun[r], halfmax(p[r]));
      const float alpha = __expf(mrun[r] - mnew);
      const float e     = __expf(p[r] - mnew);
      lrun[r] = lrun[r] * alpha + halfsum(e);
      mrun[r] = mnew;
      p[r] = e;
#pragma unroll
      for (int f = 0; f < 8; ++f) o[f][r] *= alpha;
    }

    // transpose P (C layout -> A layout) via per-wave LDS
#pragma unroll
    for (int r = 0; r < 8; ++r)
      pbuf[wave][(r + 8 * hi) * 17 + half] = p[r];
    asm volatile("s_wait_dscnt 0" ::: "memory");

    // O += P * V
#pragma unroll
    for (int kc = 0; kc < 4; ++kc) {
      v2f a;
      a.x = pbuf[wave][half * 17 + 4 * kc + 2 * hi];
      a.y = pbuf[wave][half * 17 + 4 * kc + 2 * hi + 1];
      const float* vrow = Vp + (long)(kBase + 4 * kc + 2 * hi) * (NKV * HD);
#pragma unroll
      for (int f = 0; f < 8; ++f) {
        v2f bf;
        bf.x = vrow[16 * f + half];
        bf.y = vrow[(NKV * HD) + 16 * f + half];
        o[f] = wmma_f32(a, bf, o[f]);
      }
    }
    asm volatile("s_wait_dscnt 0" ::: "memory");
  }

#pragma unroll
  for (int f = 0; f < 8; ++f) {
#pragma unroll
    for (int r = 0; r < 8; ++r) {
      const int row = qBase + r + 8 * hi;
      O[((long)b * SEQ + row) * (NHEADS * HD) + h * HD + 16 * f + half] =
          o[f][r] / lrun[r];
    }
  }
}

// ---------------------------------------------------------------------------
extern "C" void kernel_launch(void* const* d_in, const int* in_sizes, int n_in,
                              void* d_out, int out_size, void* d_ws, size_t ws_size,
                              hipStream_t stream) {
  const float* x  = (const float*)d_in[0];
  const float* wq = (const float*)d_in[1];
  const float* wk = (const float*)d_in[2];
  const float* wv = (const float*)d_in[3];
  const float* wo = (const float*)d_in[4];
  float* out = (float*)d_out;

  float* ws = (float*)d_ws;
  float* Qb = ws;                            // 4096*4096
  float* Kb = Qb + (long)MROWS * DIM;        // 4096*1024
  float* Vb = Kb + (long)MROWS * (NKV * HD); // 4096*1024
  float* AO = Vb + (long)MROWS * (NKV * HD); // 4096*4096

  const dim3 blk(256);

  gemm_wmma_f32<<<dim3(DIM / BN, MROWS / BM), blk, 0, stream>>>(
      x, wq, Qb, MROWS, DIM, DIM);
  gemm_wmma_f32<<<dim3((NKV * HD) / BN, MROWS / BM), blk, 0, stream>>>(
      x, wk, Kb, MROWS, NKV * HD, DIM);
  gemm_wmma_f32<<<dim3((NKV * HD) / BN, MROWS / BM), blk, 0, stream>>>(
      x, wv, Vb, MROWS, NKV * HD, DIM);

  {
    long totq = (long)MROWS * NHEADS * (HD / 2);
    long totk = (long)MROWS * NKV * (HD / 2);
    rope_kernel<<<(int)((totq + 255) / 256), 256, 0, stream>>>(Qb, NHEADS, totq);
    rope_kernel<<<(int)((totk + 255) / 256), 256, 0, stream>>>(Kb, NKV, totk);
  }

  attn_fa<<<dim3(1024), blk, 0, stream>>>(Qb, Kb, Vb, AO);

  gemm_wmma_f32<<<dim3(DIM / BN, MROWS / BM), blk, 0, stream>>>(
      AO, wo, out, MROWS, DIM, DIM);
}